// ItemFeat_30150670418291
// MI455X (gfx1250) — compile-verified
//
#include <hip/hip_runtime.h>

// CDNA5 / gfx1250 wave32 WMMA kernel:
// out[N,128] = mask * relu( concat(item[id], cat[cmap[id]], brand[bmap[id]]) @ W + b )
// One wave computes a 16x128 output tile via V_WMMA_F32_16X16X4_F32.

typedef __attribute__((ext_vector_type(2))) float v2f;
typedef __attribute__((ext_vector_type(8))) float v8f;

#define IN_DIM   256
#define OUT_DIM  128
#define AS       260            // padded LDS row stride (dwords): conflict-free, 16B aligned
#define WSTRIDE  260            // padded LDS stride for transposed W
#define WPB      8              // waves per block
#define TPW      4              // row-tiles per wave
#define W_FLOATS (OUT_DIM * WSTRIDE)   // 33280 floats = 133,120 B
#define A_FLOATS (16 * AS)             // 4160 floats  =  16,640 B per wave

__global__ __launch_bounds__(WPB * 32, 1)
void itemfeat_wmma(const int*   __restrict__ sample,
                   const float* __restrict__ item_emb,
                   const float* __restrict__ cat_emb,
                   const float* __restrict__ brand_emb,
                   const int*   __restrict__ cmap,
                   const int*   __restrict__ bmap,
                   const float* __restrict__ Wg,
                   const float* __restrict__ bias,
                   float*       __restrict__ out,
                   int ntiles)
{
  extern __shared__ float smem[];
  float* Wsh = smem;                                            // Wt[n][k], stride WSTRIDE
  float* Aw  = smem + W_FLOATS + (threadIdx.x >> 5) * A_FLOATS; // per-wave A tile

  // Stage W transposed into LDS once per block: Wsh[n*WSTRIDE + k] = W[k*128 + n]
  for (int idx = threadIdx.x; idx < IN_DIM * OUT_DIM; idx += blockDim.x) {
    int k = idx >> 7;
    int n = idx & 127;
    Wsh[n * WSTRIDE + k] = Wg[idx];
  }
  __syncthreads();

  const int lane = threadIdx.x & 31;
  const int nloc = lane & 15;     // column within a 16-wide tile / row within A tile
  const int hsel = lane >> 4;     // wave half
  const int koff = hsel * 2;      // f32 16x16x4 layout: upper half holds K+2,K+3

  // 8 column-tile B-fragment base pointers into transposed W
  const float* Bb[8];
#pragma unroll
  for (int c = 0; c < 8; ++c)
    Bb[c] = Wsh + (c * 16 + nloc) * WSTRIDE + koff;

  const int gwave = blockIdx.x * WPB + (threadIdx.x >> 5);

  for (int t = 0; t < TPW; ++t) {
    int tile = gwave * TPW + t;
    if (tile >= ntiles) return;           // no barriers below: divergent exit is safe
    int rowBase = tile * 16;

    // ---- Gather: 16 rows x 256 f32 features -> LDS (coalesced float4 per table row)
    for (int r = 0; r < 16; ++r) {
      int sid = sample[rowBase + r];      // uniform across wave
      int cid = cmap[sid];
      int bid = bmap[sid];
      // item embedding: 128 floats = 32 x float4, one per lane
      const float4 idv = *(const float4*)(item_emb + (size_t)sid * 128 + lane * 4);
      // category (lanes 0-15) + brand (lanes 16-31): 64+64 floats
      const float* attr = (lane < 16)
          ? (cat_emb   + (size_t)cid * 64 + lane * 4)
          : (brand_emb + (size_t)bid * 64 + (lane - 16) * 4);
      const float4 atv = *(const float4*)attr;
      *(float4*)(Aw + r * AS + lane * 4)       = idv;   // ds_store_b128
      *(float4*)(Aw + r * AS + 128 + lane * 4) = atv;   // ds_store_b128
    }
    // same-wave LDS ordering: stores precede the fragment loads below

    // ---- 16x128 = sum_k A(16x4) * B(4x16) via V_WMMA_F32_16X16X4_F32
    v8f acc[8] = {};
    const float* Ar = Aw + nloc * AS + koff;
#pragma unroll 4
    for (int k0 = 0; k0 < IN_DIM; k0 += 4) {
      v2f a = *(const v2f*)(Ar + k0);                   // ds_load_b64
#pragma unroll
      for (int c = 0; c < 8; ++c) {
        v2f b = *(const v2f*)(Bb[c] + k0);              // ds_load_b64
        acc[c] = __builtin_amdgcn_wmma_f32_16x16x4_f32(
            /*neg_a=*/false, a, /*neg_b=*/false, b,
            /*c_mod=*/(short)0, acc[c],
            /*reuse_a=*/false, /*reuse_b=*/false);
      }
    }

    // ---- Epilogue: bias + ReLU + pad-row mask, store 16x128 tile
    // C/D layout: lane half h, vgpr j -> row (8h + j), col = c*16 + (lane&15)
    int rhalf = hsel * 8;
    unsigned mask = 0;
#pragma unroll
    for (int j = 0; j < 8; ++j)
      if (sample[rowBase + rhalf + j] != 0) mask |= (1u << j);

#pragma unroll
    for (int c = 0; c < 8; ++c) {
      int col = c * 16 + nloc;
      float bb = bias[col];
#pragma unroll
      for (int j = 0; j < 8; ++j) {
        float v = acc[c][j] + bb;
        v = v > 0.0f ? v : 0.0f;
        if (!(mask & (1u << j))) v = 0.0f;
        out[(size_t)(rowBase + rhalf + j) * OUT_DIM + col] = v;
      }
    }
  }
}

extern "C" void kernel_launch(void* const* d_in, const int* in_sizes, int n_in,
                              void* d_out, int out_size, void* d_ws, size_t ws_size,
                              hipStream_t stream) {
  const int*   sample    = (const int*)  d_in[0];
  const float* item_emb  = (const float*)d_in[1];
  const float* cat_emb   = (const float*)d_in[2];
  const float* brand_emb = (const float*)d_in[3];
  const int*   cmap      = (const int*)  d_in[4];
  const int*   bmap      = (const int*)  d_in[5];
  const float* W         = (const float*)d_in[6];
  const float* b         = (const float*)d_in[7];
  float*       out       = (float*)d_out;

  int N = in_sizes[0];            // B*L = 409600 (divisible by 16)
  int ntiles = N / 16;
  int wavesNeeded = (ntiles + TPW - 1) / TPW;
  int blocks = (wavesNeeded + WPB - 1) / WPB;
  size_t smemBytes = (size_t)(W_FLOATS + WPB * A_FLOATS) * sizeof(float); // ~260 KB <= 320 KB/WGP

  itemfeat_wmma<<<blocks, WPB * 32, smemBytes, stream>>>(
      sample, item_emb, cat_emb, brand_emb, cmap, bmap, W, b, out, ntiles);
}